// GraphAttentionNetwork_57612691309001
// MI455X (gfx1250) — compile-verified
//
#include <hip/hip_runtime.h>
#include <math.h>

typedef __attribute__((ext_vector_type(2))) float v2f;
typedef __attribute__((ext_vector_type(8))) float v8f;

#define N_NODES   50000     // == 3125 * 16, exactly divisible by 16
#define E_EDGES   800000
#define IN_CH_    128
#define OUT_CH_   32
#define HEADS_    10
#define HC1_      320          // HEADS*OUT_CH
#define NEG_SLOPE_ 0.2f
#define NUM_GRAPHS_ 512

// ---------------------------------------------------------------- utilities

__global__ void fill_kernel(float* __restrict__ p, float v, int n) {
  int i = blockIdx.x * blockDim.x + threadIdx.x;
  if (i < n) p[i] = v;
}

__device__ __forceinline__ void atomic_max_f32(float* addr, float val) {
  // valid for slots initialized to -inf: non-negative floats order as signed
  // ints, negative floats order inversely as unsigned ints.
  if (val >= 0.0f)
    atomicMax((int*)addr, __float_as_int(val));
  else
    atomicMin((unsigned int*)addr, (unsigned int)__float_as_int(val));
}

// ------------------------------------------------- FP32 WMMA GEMM (16x16x4)
// D[M,NCOLS] = A[M,K] @ B[K,NCOLS], row-major f32. M must be a multiple of 16.
// grid = (M/16, NCOLS/(16*NT)), block = 32 (one wave, EXEC all ones, no
// divergence anywhere). Each wave produces a 16 x (16*NT) strip, reusing its
// A fragment NT times per K-step.
//
// Fragment layouts (ISA 7.12.2, 32-bit operands):
//   A 16x4 : lanes 0-15 rows 0-15 K={0,1}; lanes 16-31 rows 0-15 K={2,3}
//            -> per-lane contiguous pair, loaded as one b64.
//   B 4x16 : lane half selects K pair, l16 selects column.
//   C/D    : VGPR r -> row (r + 8*half), lane l16 -> column.
template <int K, int NCOLS, int NT>
__global__ void gemm_wmma_f32_kernel(const float* __restrict__ A,
                                     const float* __restrict__ B,
                                     float* __restrict__ D) {
  const int lane = threadIdx.x;
  const int half = lane >> 4;       // 0: lanes 0-15, 1: lanes 16-31
  const int l16  = lane & 15;
  const int row  = blockIdx.x * 16 + l16;
  const int col0 = blockIdx.y * (16 * NT) + l16;

  const float* ap = A + (size_t)row * K + half * 2;     // A[row][half*2 + k]
  const float* bp = B + (size_t)(half * 2) * NCOLS + col0;

  v8f acc[NT];
#pragma unroll
  for (int t = 0; t < NT; ++t) acc[t] = (v8f){};

#pragma unroll
  for (int k = 0; k < K; k += 4) {
    const v2f a = *(const v2f*)(ap + k);                // one global_load_b64
#pragma unroll
    for (int t = 0; t < NT; ++t) {
      v2f b;
      b.x = bp[(size_t)k * NCOLS + t * 16];             // constant offsets
      b.y = bp[(size_t)(k + 1) * NCOLS + t * 16];
      acc[t] = __builtin_amdgcn_wmma_f32_16x16x4_f32(
          /*neg_a=*/false, a, /*neg_b=*/false, b,
          /*c_mod=*/(short)0, acc[t], /*reuse_a=*/false, /*reuse_b=*/false);
    }
  }

  const int mbase = blockIdx.x * 16 + half * 8;
#pragma unroll
  for (int t = 0; t < NT; ++t) {
    float* dp = D + (size_t)mbase * NCOLS + col0 + t * 16;
#pragma unroll
    for (int r = 0; r < 8; ++r)
      dp[(size_t)r * NCOLS] = acc[t][r];
  }
}

// ----------------------------------------- per-node attention scores e_src/e_dst
__global__ void node_attn_e_kernel(const float* __restrict__ h,
                                   const float* __restrict__ a_src,
                                   const float* __restrict__ a_dst,
                                   float* __restrict__ e_src,
                                   float* __restrict__ e_dst,
                                   int nH, int H, int C) {
  int i = blockIdx.x * blockDim.x + threadIdx.x;
  if (i >= nH) return;
  const int hd = i % H;
  const float* hp = h + (size_t)i * C;
  const float* as = a_src + (size_t)hd * C;
  const float* ad = a_dst + (size_t)hd * C;
  float es = 0.0f, ed = 0.0f;
  for (int c = 0; c < C; ++c) {
    float v = hp[c];
    es = fmaf(v, as[c], es);
    ed = fmaf(v, ad[c], ed);
  }
  e_src[i] = es;
  e_dst[i] = ed;
}

// ---------------------------------------------------------- edge pass A: alpha+max
__global__ void edge_alpha_kernel(const int* __restrict__ src,
                                  const int* __restrict__ dst,
                                  const float* __restrict__ e_src,
                                  const float* __restrict__ e_dst,
                                  float* __restrict__ alpha,
                                  float* __restrict__ amax,
                                  int EH, int H) {
  int i = blockIdx.x * blockDim.x + threadIdx.x;
  if (i >= EH) return;
  const int e = i / H, hd = i % H;
  float a = e_src[src[e] * H + hd] + e_dst[dst[e] * H + hd];
  a = (a > 0.0f) ? a : NEG_SLOPE_ * a;       // leaky_relu
  alpha[i] = a;
  atomic_max_f32(&amax[dst[e] * H + hd], a);
}

// -------------------------------------------------------- edge pass B: exp + denom
__global__ void edge_exp_kernel(const int* __restrict__ dst,
                                const float* __restrict__ amax,
                                float* __restrict__ alpha_ex,   // in: alpha, out: exp
                                float* __restrict__ denom,
                                int EH, int H) {
  int i = blockIdx.x * blockDim.x + threadIdx.x;
  if (i >= EH) return;
  const int e = i / H, hd = i % H;
  const float m = amax[dst[e] * H + hd];     // finite: this dst has >=1 in-edge
  const float ex = expf(alpha_ex[i] - m);
  alpha_ex[i] = ex;
  atomicAdd(&denom[dst[e] * H + hd], ex);
}

// ------------------------------------------------- edge pass C: message aggregation
__global__ void edge_msg_kernel(const int* __restrict__ src,
                                const int* __restrict__ dst,
                                const float* __restrict__ ex,
                                const float* __restrict__ denom,
                                const float* __restrict__ h,
                                float* __restrict__ agg,
                                int EH, int H, int C) {
  int i = blockIdx.x * blockDim.x + threadIdx.x;
  if (i >= EH) return;
  const int e = i / H, hd = i % H;
  const float coef = ex[i] / (denom[dst[e] * H + hd] + 1e-16f);
  const float* hp = h + ((size_t)src[e] * H + hd) * C;
  float* op = agg + ((size_t)dst[e] * H + hd) * C;
  for (int c = 0; c < C; ++c)
    atomicAdd(&op[c], hp[c] * coef);
}

// ---------------------------------------------------------------- bias + ELU
__global__ void elu_bias_kernel(float* __restrict__ x,
                                const float* __restrict__ b,
                                int total, int F) {
  int i = blockIdx.x * blockDim.x + threadIdx.x;
  if (i >= total) return;
  float v = x[i] + b[i % F];
  x[i] = (v > 0.0f) ? v : expm1f(v);
}

// ---------------------------------------------------------------- global max pool
__global__ void pool_max_kernel(const float* __restrict__ h,
                                const int* __restrict__ batch,
                                float* __restrict__ pooled,
                                int total) {
  int i = blockIdx.x * blockDim.x + threadIdx.x;
  if (i >= total) return;
  const int n = i >> 5, c = i & 31;
  atomic_max_f32(&pooled[batch[n] * 32 + c], h[i]);
}

// ------------------------------------------------------------- final FC + relu
__global__ void final_fc_kernel(const float* __restrict__ pooled,
                                const float* __restrict__ fc_w,
                                const float* __restrict__ fc_b,
                                float* __restrict__ out,
                                int total) {
  int i = blockIdx.x * blockDim.x + threadIdx.x;
  if (i >= total) return;
  const int g = i >> 5, j = i & 31;
  float acc = fc_b[j];
  for (int c = 0; c < 32; ++c) {
    float p = pooled[g * 32 + c];
    p = (p == -INFINITY) ? 0.0f : p;   // empty segments -> 0
    acc = fmaf(p, fc_w[c * 32 + j], acc);
  }
  out[i] = fmaxf(acc, 0.0f);
}

// ---------------------------------------------------------------------------

extern "C" void kernel_launch(void* const* d_in, const int* in_sizes, int n_in,
                              void* d_out, int out_size, void* d_ws, size_t ws_size,
                              hipStream_t stream) {
  (void)in_sizes; (void)n_in; (void)out_size; (void)ws_size;

  const float* x      = (const float*)d_in[0];
  const int*   eidx   = (const int*)  d_in[1];
  const int*   batch  = (const int*)  d_in[2];
  /* d_in[3] = num_graphs scalar (== 512, compile-time constant here) */
  const float* W1     = (const float*)d_in[4];
  const float* a_src1 = (const float*)d_in[5];
  const float* a_dst1 = (const float*)d_in[6];
  const float* b1     = (const float*)d_in[7];
  const float* W2     = (const float*)d_in[8];
  const float* a_src2 = (const float*)d_in[9];
  const float* a_dst2 = (const float*)d_in[10];
  const float* b2     = (const float*)d_in[11];
  const float* fc_w   = (const float*)d_in[12];
  const float* fc_b   = (const float*)d_in[13];

  const int* src = eidx;
  const int* dst = eidx + E_EDGES;
  float* out = (float*)d_out;

  // ---- workspace carve-out (floats) ----
  float* ws = (float*)d_ws;
  size_t off = 0;
  float* h1     = ws + off; off += (size_t)N_NODES * HC1_;     // 16.0M
  float* agg1   = ws + off; off += (size_t)N_NODES * HC1_;     // 16.0M
  float* es1    = ws + off; off += (size_t)N_NODES * HEADS_;
  float* ed1    = ws + off; off += (size_t)N_NODES * HEADS_;
  float* amax1  = ws + off; off += (size_t)N_NODES * HEADS_;
  float* denom1 = ws + off; off += (size_t)N_NODES * HEADS_;
  float* ex1    = ws + off; off += (size_t)E_EDGES * HEADS_;   // 8.0M
  float* h2     = ws + off; off += (size_t)N_NODES * OUT_CH_;
  float* agg2   = ws + off; off += (size_t)N_NODES * OUT_CH_;
  float* es2    = ws + off; off += (size_t)N_NODES;
  float* ed2    = ws + off; off += (size_t)N_NODES;
  float* amax2  = ws + off; off += (size_t)N_NODES;
  float* denom2 = ws + off; off += (size_t)N_NODES;
  float* ex2    = ws + off; off += (size_t)E_EDGES;
  float* pooled = ws + off; off += (size_t)NUM_GRAPHS_ * OUT_CH_;

  const int nh1   = N_NODES * HEADS_;
  const int eh1   = E_EDGES * HEADS_;
  const int nf1   = N_NODES * HC1_;
  const int nf2   = N_NODES * OUT_CH_;
  const int pool1 = NUM_GRAPHS_ * OUT_CH_;
  const int B = 256;
  #define GRID(n) dim3(((n) + B - 1) / B)

  // ================= layer 1 =================
  // [50000,128] @ [128,320]; each wave: 16x64 strip -> grid (3125, 5)
  gemm_wmma_f32_kernel<IN_CH_, HC1_, 4>
      <<<dim3(N_NODES / 16, HC1_ / 64), 32, 0, stream>>>(x, W1, h1);

  node_attn_e_kernel<<<GRID(nh1), B, 0, stream>>>(h1, a_src1, a_dst1, es1, ed1,
                                                  nh1, HEADS_, OUT_CH_);
  fill_kernel<<<GRID(nh1), B, 0, stream>>>(amax1, -INFINITY, nh1);
  fill_kernel<<<GRID(nh1), B, 0, stream>>>(denom1, 0.0f, nh1);
  fill_kernel<<<GRID(nf1), B, 0, stream>>>(agg1, 0.0f, nf1);

  edge_alpha_kernel<<<GRID(eh1), B, 0, stream>>>(src, dst, es1, ed1, ex1, amax1,
                                                 eh1, HEADS_);
  edge_exp_kernel<<<GRID(eh1), B, 0, stream>>>(dst, amax1, ex1, denom1, eh1, HEADS_);
  edge_msg_kernel<<<GRID(eh1), B, 0, stream>>>(src, dst, ex1, denom1, h1, agg1,
                                               eh1, HEADS_, OUT_CH_);
  elu_bias_kernel<<<GRID(nf1), B, 0, stream>>>(agg1, b1, nf1, HC1_);

  // ================= layer 2 =================
  // [50000,320] @ [320,32]; each wave: 16x32 strip -> grid (3125, 1)
  gemm_wmma_f32_kernel<HC1_, OUT_CH_, 2>
      <<<dim3(N_NODES / 16, OUT_CH_ / 32), 32, 0, stream>>>(agg1, W2, h2);

  node_attn_e_kernel<<<GRID(N_NODES), B, 0, stream>>>(h2, a_src2, a_dst2, es2, ed2,
                                                      N_NODES, 1, OUT_CH_);
  fill_kernel<<<GRID(N_NODES), B, 0, stream>>>(amax2, -INFINITY, N_NODES);
  fill_kernel<<<GRID(N_NODES), B, 0, stream>>>(denom2, 0.0f, N_NODES);
  fill_kernel<<<GRID(nf2), B, 0, stream>>>(agg2, 0.0f, nf2);

  edge_alpha_kernel<<<GRID(E_EDGES), B, 0, stream>>>(src, dst, es2, ed2, ex2, amax2,
                                                     E_EDGES, 1);
  edge_exp_kernel<<<GRID(E_EDGES), B, 0, stream>>>(dst, amax2, ex2, denom2,
                                                   E_EDGES, 1);
  edge_msg_kernel<<<GRID(E_EDGES), B, 0, stream>>>(src, dst, ex2, denom2, h2, agg2,
                                                   E_EDGES, 1, OUT_CH_);
  elu_bias_kernel<<<GRID(nf2), B, 0, stream>>>(agg2, b2, nf2, OUT_CH_);

  // ================= pool + FC =================
  fill_kernel<<<GRID(pool1), B, 0, stream>>>(pooled, -INFINITY, pool1);
  pool_max_kernel<<<GRID(nf2), B, 0, stream>>>(agg2, batch, pooled, nf2);
  final_fc_kernel<<<GRID(pool1), B, 0, stream>>>(pooled, fc_w, fc_b, out, pool1);

  #undef GRID
}